// KGEA_28106265985638
// MI455X (gfx1250) — compile-verified
//
#include <hip/hip_runtime.h>
#include <math.h>

#define DIM 128

typedef __attribute__((ext_vector_type(2))) float v2f;
typedef __attribute__((ext_vector_type(8))) float v8f;

// ---------- helpers ----------
__device__ __forceinline__ unsigned fflip(float f) {
  unsigned u = __float_as_uint(f);
  return u ^ ((u & 0x80000000u) ? 0xFFFFFFFFu : 0x80000000u);
}
__device__ __forceinline__ float funflip(unsigned u) {
  return __uint_as_float(u ^ ((u & 0x80000000u) ? 0x80000000u : 0xFFFFFFFFu));
}
__device__ __forceinline__ float leaky(float x) { return x >= 0.f ? x : 0.2f * x; }

// ---------- LayerNorm: one wave32 per row of [M,128] ----------
__global__ __launch_bounds__(256) void ln_kernel(const float* __restrict__ src,
                                                 const float* __restrict__ g,
                                                 const float* __restrict__ b,
                                                 float* __restrict__ dst, int M) {
  int row = blockIdx.x * 8 + (threadIdx.x >> 5);
  if (row >= M) return;
  int lane = threadIdx.x & 31;
  const float4 x = reinterpret_cast<const float4*>(src + (size_t)row * DIM)[lane];
  float s = x.x + x.y + x.z + x.w;
  float q = x.x * x.x + x.y * x.y + x.z * x.z + x.w * x.w;
#pragma unroll
  for (int off = 16; off > 0; off >>= 1) {
    s += __shfl_xor(s, off, 32);
    q += __shfl_xor(q, off, 32);
  }
  float mean = s * (1.0f / DIM);
  float var = q * (1.0f / DIM) - mean * mean;
  float rstd = rsqrtf(var + 1e-6f);
  const float4 gg = reinterpret_cast<const float4*>(g)[lane];
  const float4 bb = reinterpret_cast<const float4*>(b)[lane];
  float4 y;
  y.x = (x.x - mean) * rstd * gg.x + bb.x;
  y.y = (x.y - mean) * rstd * gg.y + bb.y;
  y.z = (x.z - mean) * rstd * gg.z + bb.z;
  y.w = (x.w - mean) * rstd * gg.w + bb.w;
  reinterpret_cast<float4*>(dst + (size_t)row * DIM)[lane] = y;
}

// ---------- GEMM C[M,128] = A[M,128] @ W[128,128] + bias, fp32 WMMA ----------
// Block: 128 threads = 4 wave32, 64 rows per block.
// W staged in LDS in K-pair-interleaved layout:
//   lW[(k>>1)*256 + c*2 + (k&1)] = W[k*128 + c]
// so each B fragment (two consecutive K for one col) is one aligned ds_read_b64.
// V_WMMA_F32_16X16X4_F32 fragment layouts (ISA 7.12.2):
//   A 16x4:  lane m=lane%16 holds K = 2*(lane/16)+{0,1}
//   B 4x16:  lane n=lane%16 holds K = 2*(lane/16)+{0,1}  (mirrored)
//   C 16x16: VGPR v holds row M = v + 8*(lane/16), col N = lane%16
__global__ __launch_bounds__(128) void gemm128_wmma_kernel(
    const float* __restrict__ A, const float* __restrict__ W,
    const float* __restrict__ bias, float* __restrict__ C, int M) {
  __shared__ float lW[DIM * DIM];
  int tid = threadIdx.x;
  for (int i = tid; i < DIM * DIM / 4; i += 128) {
    int k = i >> 5;            // W row
    int c4 = (i & 31) * 4;     // W col start
    float4 wv = reinterpret_cast<const float4*>(W)[i];
    float* base = lW + ((k >> 1) * 256) + (k & 1);
    base[(c4 + 0) * 2] = wv.x;
    base[(c4 + 1) * 2] = wv.y;
    base[(c4 + 2) * 2] = wv.z;
    base[(c4 + 3) * 2] = wv.w;
  }
  __syncthreads();

  int wave = tid >> 5, lane = tid & 31;
  int rowBase = blockIdx.x * 64 + wave * 16;
  int mrow = lane & 15;   // M (for A) and N (for B/C)
  int khalf = lane >> 4;  // 0/1 -> K offset 0/2
  int row = rowBase + mrow;
  int rowC = row < M ? row : (M - 1);
  const float* Arow = A + (size_t)rowC * DIM;

  v8f acc[8];
#pragma unroll
  for (int t = 0; t < 8; ++t) {
    float bv = bias[t * 16 + mrow];
#pragma unroll
    for (int v = 0; v < 8; ++v) acc[t][v] = bv;
  }

  for (int k0 = 0; k0 < DIM; k0 += 4) {
    int ka = k0 + 2 * khalf;
    v2f af = *reinterpret_cast<const v2f*>(Arow + ka);       // 8B aligned (ka even)
    const float* wrow = lW + (ka >> 1) * 256;
#pragma unroll
    for (int t = 0; t < 8; ++t) {
      int col = t * 16 + mrow;
      v2f bf = *reinterpret_cast<const v2f*>(wrow + col * 2);  // one ds_read_b64
      acc[t] = __builtin_amdgcn_wmma_f32_16x16x4_f32(
          false, af, false, bf, (short)0, acc[t], false, false);
    }
  }

  // Store: wave-uniform fast path (full 16-row strip) avoids per-store EXEC churn.
  if (rowBase + 16 <= M) {
#pragma unroll
    for (int t = 0; t < 8; ++t) {
      int col = t * 16 + mrow;
#pragma unroll
      for (int v = 0; v < 8; ++v) {
        int r = rowBase + v + 8 * khalf;
        C[(size_t)r * DIM + col] = acc[t][v];
      }
    }
  } else {
#pragma unroll
    for (int t = 0; t < 8; ++t) {
      int col = t * 16 + mrow;
#pragma unroll
      for (int v = 0; v < 8; ++v) {
        int r = rowBase + v + 8 * khalf;
        if (r < M) C[(size_t)r * DIM + col] = acc[t][v];
      }
    }
  }
}

// ---------- s[m] = tanh(dot(A[m], B[m])): wave per row ----------
__global__ __launch_bounds__(256) void rowdot_tanh_kernel(
    const float* __restrict__ A, const float* __restrict__ B, float* __restrict__ s, int M) {
  int row = blockIdx.x * 8 + (threadIdx.x >> 5);
  if (row >= M) return;
  int lane = threadIdx.x & 31;
  float4 a = reinterpret_cast<const float4*>(A + (size_t)row * DIM)[lane];
  float4 b = reinterpret_cast<const float4*>(B + (size_t)row * DIM)[lane];
  float d = a.x * b.x + a.y * b.y + a.z * b.z + a.w * b.w;
#pragma unroll
  for (int off = 16; off > 0; off >>= 1) d += __shfl_xor(d, off, 32);
  if (lane == 0) s[row] = tanhf(d);
}

// ---------- s[m] = tanh(X[m]·wrs[0:128] + wrs[128]): wave per row ----------
__global__ __launch_bounds__(256) void gemv_tanh_kernel(
    const float* __restrict__ X, const float* __restrict__ wrs, float* __restrict__ s, int M) {
  int row = blockIdx.x * 8 + (threadIdx.x >> 5);
  if (row >= M) return;
  int lane = threadIdx.x & 31;
  float4 x = reinterpret_cast<const float4*>(X + (size_t)row * DIM)[lane];
  float4 wv = reinterpret_cast<const float4*>(wrs)[lane];
  float d = x.x * wv.x + x.y * wv.y + x.z * wv.z + x.w * wv.w;
#pragma unroll
  for (int off = 16; off > 0; off >>= 1) d += __shfl_xor(d, off, 32);
  if (lane == 0) s[row] = tanhf(d + wrs[DIM]);
}

// ---------- wrs[k] = sum_c W[k,c]; wrs[128] = sum(b) ----------
__global__ void rowsum_w_kernel(const float* __restrict__ W, const float* __restrict__ b,
                                float* __restrict__ wrs) {
  int t = threadIdx.x;
  if (t < DIM) {
    float s = 0.f;
    for (int c = 0; c < DIM; ++c) s += W[t * DIM + c];
    wrs[t] = s;
  } else if (t == DIM) {
    float s = 0.f;
    for (int c = 0; c < DIM; ++c) s += b[c];
    wrs[DIM] = s;
  }
}

// ---------- segment softmax + SpMM ----------
__global__ void seg_init_kernel(float* __restrict__ agg, unsigned* __restrict__ rowmax,
                                float* __restrict__ rowsum, int n, size_t n128) {
  size_t i = (size_t)blockIdx.x * blockDim.x + threadIdx.x;
  if (i < n128) agg[i] = 0.f;
  if (i < (size_t)n) {
    rowmax[i] = fflip(-3.402823466e38f);
    rowsum[i] = 0.f;
  }
}

__global__ void edge_logits_kernel(const float* __restrict__ val, const int* __restrict__ row,
                                   const int* __restrict__ col, const float* __restrict__ s1,
                                   const float* __restrict__ s2, float* __restrict__ elog,
                                   unsigned* __restrict__ rowmax, int ne) {
  int e = blockIdx.x * blockDim.x + threadIdx.x;
  if (e >= ne) return;
  int r = row[e];
  float l;
  if (s1)
    l = leaky(val[e] * (s1[r] + s2[col[e]]));
  else
    l = leaky(2.0f * val[e]);
  elog[e] = l;
  atomicMax(&rowmax[r], fflip(l));
}

__global__ void edge_exp_kernel(float* __restrict__ elog, const int* __restrict__ row,
                                const unsigned* __restrict__ rowmax, float* __restrict__ rowsum,
                                int ne) {
  int e = blockIdx.x * blockDim.x + threadIdx.x;
  if (e >= ne) return;
  int r = row[e];
  float w = expf(elog[e] - funflip(rowmax[r]));
  elog[e] = w;
  atomicAdd(&rowsum[r], w);
}

// wave per edge: agg[row] += (w/sum[row]) * V[col]
__global__ __launch_bounds__(256) void edge_spmm_kernel(
    const float* __restrict__ elog, const float* __restrict__ rowsum,
    const int* __restrict__ row, const int* __restrict__ col, const float* __restrict__ V,
    float* __restrict__ agg, int ne) {
  int e = blockIdx.x * 8 + (threadIdx.x >> 5);
  if (e >= ne) return;
  int lane = threadIdx.x & 31;
  int r = row[e], c = col[e];
  float a = elog[e] / rowsum[r];
  float4 v = reinterpret_cast<const float4*>(V + (size_t)c * DIM)[lane];
  float* dst = agg + (size_t)r * DIM + lane * 4;
  atomicAdd(dst + 0, a * v.x);
  atomicAdd(dst + 1, a * v.y);
  atomicAdd(dst + 2, a * v.z);
  atomicAdd(dst + 3, a * v.w);
}

// ---------- elementwise combines ----------
__global__ void combine2_kernel(float* __restrict__ dst, const float* __restrict__ a,
                                const float* __restrict__ b, size_t n) {
  size_t i = (size_t)blockIdx.x * blockDim.x + threadIdx.x;
  if (i < n) dst[i] = a[i] + 0.6f * b[i];
}
__global__ void combine3_kernel(float* __restrict__ dst, const float* __restrict__ o1,
                                const float* __restrict__ a, const float* __restrict__ b,
                                size_t n) {
  size_t i = (size_t)blockIdx.x * blockDim.x + threadIdx.x;
  if (i < n) dst[i] = 2.0f * o1[i] + a[i] + 0.6f * b[i];
}

// ---------- launch ----------
extern "C" void kernel_launch(void* const* d_in, const int* in_sizes, int n_in,
                              void* d_out, int out_size, void* d_ws, size_t ws_size,
                              hipStream_t stream) {
  const float* ent = (const float*)d_in[0];
  const float* rel = (const float*)d_in[1];
  const int* e_row = (const int*)d_in[2];
  const int* e_col = (const int*)d_in[3];
  const float* e_val = (const float*)d_in[4];
  const int* r_row = (const int*)d_in[5];
  const int* r_col = (const int*)d_in[6];
  const float* r_val = (const float*)d_in[7];
  auto prm = [&](int i) { return (const float*)d_in[i]; };

  int N = in_sizes[0] / DIM;
  int R = in_sizes[1] / DIM;
  int E = in_sizes[4];
  int E2 = in_sizes[7];
  size_t N128 = (size_t)N * DIM;
  size_t R128 = (size_t)R * DIM;

  float* w = (float*)d_ws;
  float* buf0 = w; w += N128;
  float* buf1 = w; w += N128;
  float* buf2 = w; w += N128;  // always the segment-sum accumulator
  float* buf3 = w; w += N128;
  float* buf4 = w; w += N128;
  float* bufR0 = w; w += R128;
  float* bufR1 = w; w += R128;
  float* bufR2 = w; w += R128;
  float* sN = w; w += N;
  float* sN2 = w; w += N;
  float* sR = w; w += R;
  unsigned* rowmax = (unsigned*)w; w += N;
  float* rowsum = w; w += N;
  float* elog = w; w += (E > E2 ? E : E2);
  float* wrs1 = w; w += 132;
  float* wrs2 = w; w += 132;
  float* out1 = (float*)d_out;
  float* out2 = out1 + N128;

  auto LN = [&](const float* s, const float* g, const float* b, float* d, int M) {
    ln_kernel<<<(M + 7) / 8, 256, 0, stream>>>(s, g, b, d, M);
  };
  auto GEMM = [&](const float* A, const float* Wm, const float* b, float* C, int M) {
    gemm128_wmma_kernel<<<(M + 63) / 64, 128, 0, stream>>>(A, Wm, b, C, M);
  };
  auto SEGINIT = [&]() {
    seg_init_kernel<<<(int)((N128 + 255) / 256), 256, 0, stream>>>(buf2, rowmax, rowsum, N, N128);
  };
  auto ELOG = [&](const float* v, const int* r, const int* c, const float* s1, const float* s2, int ne) {
    edge_logits_kernel<<<(ne + 255) / 256, 256, 0, stream>>>(v, r, c, s1, s2, elog, rowmax, ne);
  };
  auto EEXP = [&](const int* r, int ne) {
    edge_exp_kernel<<<(ne + 255) / 256, 256, 0, stream>>>(elog, r, rowmax, rowsum, ne);
  };
  auto ESPMM = [&](const int* r, const int* c, const float* V, int ne) {
    edge_spmm_kernel<<<(ne + 7) / 8, 256, 0, stream>>>(elog, rowsum, r, c, V, buf2, ne);
  };
  auto RD = [&](const float* A, const float* B, float* s, int M) {
    rowdot_tanh_kernel<<<(M + 7) / 8, 256, 0, stream>>>(A, B, s, M);
  };

  // ============== conv0: _decent_attn (mean_sum=True, key=None) ==============
  LN(ent, prm(21), prm(20), buf0, N);        // q0 = LN(ent, c0.local.ln1)
  GEMM(buf0, prm(27), prm(26), buf1, N);     // value0 = q0 @ c0.local.w
  SEGINIT();
  ELOG(e_val, e_row, e_col, nullptr, nullptr, E);  // leaky(2*val)
  EEXP(e_row, E);
  ESPMM(e_row, e_col, buf1, E);
  LN(buf2, prm(23), prm(22), buf3, N);       // e_out = LN(agg, c0.local.ln2)

  // ============== conv0: _linear_attn (query=e_out, inp=rel) ==============
  LN(rel, prm(11), prm(10), bufR0, R);       // x = LN(rel, c0.er.ln1)
  GEMM(bufR0, prm(15), prm(14), bufR1, R);   // dns = x @ c0.er.w
  GEMM(buf3, prm(17), prm(16), buf0, N);     // T = e_out @ c0.er.w1 + b
  RD(buf0, buf3, sN, N);                     // s1 = tanh(rowdot(T, e_out))
  GEMM(bufR0, prm(19), prm(18), bufR2, R);   // TR = x @ c0.er.w2 + b
  RD(bufR2, bufR0, sR, R);                   // s2
  SEGINIT();
  ELOG(r_val, r_row, r_col, sN, sR, E2);
  EEXP(r_row, E2);
  ESPMM(r_row, r_col, bufR1, E2);
  LN(buf2, prm(13), prm(12), buf4, N);       // er = LN(agg, c0.er.ln2)
  GEMM(buf4, prm(9), prm(8), buf2, N);       // d1 = er @ c0.dense1
  combine2_kernel<<<(int)((N128 + 255) / 256), 256, 0, stream>>>(buf2, buf3, buf2, N128);
  LN(buf2, prm(57), prm(56), out1, N);       // out1 = LN(e_out + 0.6*d1, ln0)

  // ============== conv1: _decent_attn (key=ent, mean_sum=False) ==============
  LN(out1, prm(45), prm(44), buf0, N);       // q1 = LN(out1, c1.local.ln1)
  LN(ent, prm(49), prm(48), buf1, N);        // k1 = LN(ent, c1.local.lnk)
  GEMM(buf1, prm(51), prm(50), buf3, N);     // value1 = k1 @ c1.local.w
  // sum(dense(x,W),axis=1) == x·(W@1) + sum(b)  -> GEMV
  rowsum_w_kernel<<<1, 256, 0, stream>>>(prm(53), prm(52), wrs1);
  gemv_tanh_kernel<<<(N + 7) / 8, 256, 0, stream>>>(buf0, wrs1, sN, N);   // s1(q1)
  rowsum_w_kernel<<<1, 256, 0, stream>>>(prm(55), prm(54), wrs2);
  gemv_tanh_kernel<<<(N + 7) / 8, 256, 0, stream>>>(buf1, wrs2, sN2, N);  // s2(k1)
  SEGINIT();
  ELOG(e_val, e_row, e_col, sN, sN2, E);
  EEXP(e_row, E);
  ESPMM(e_row, e_col, buf3, E);
  LN(buf2, prm(47), prm(46), buf4, N);       // e_out2 = LN(agg, c1.local.ln2)

  // ============== conv1: _linear_attn (query=e_out2, inp=rel) ==============
  LN(rel, prm(35), prm(34), bufR0, R);
  GEMM(bufR0, prm(39), prm(38), bufR1, R);   // dns1
  GEMM(buf4, prm(41), prm(40), buf0, N);     // T = e_out2 @ c1.er.w1 + b
  RD(buf0, buf4, sN, N);
  GEMM(bufR0, prm(43), prm(42), bufR2, R);
  RD(bufR2, bufR0, sR, R);
  SEGINIT();
  ELOG(r_val, r_row, r_col, sN, sR, E2);
  EEXP(r_row, E2);
  ESPMM(r_row, r_col, bufR1, E2);
  LN(buf2, prm(37), prm(36), buf1, N);       // er2
  GEMM(buf1, prm(33), prm(32), buf0, N);     // d2 = er2 @ c1.dense1
  combine3_kernel<<<(int)((N128 + 255) / 256), 256, 0, stream>>>(buf2, out1, buf4, buf0, N128);
  LN(buf2, prm(59), prm(58), out2, N);       // out2 = LN(2*out1 + e_out2 + 0.6*d2, ln1)
}